// Attention_48266842472985
// MI455X (gfx1250) — compile-verified
//
#include <hip/hip_runtime.h>
#include <math.h>

typedef float v2f __attribute__((ext_vector_type(2)));
typedef float v4f __attribute__((ext_vector_type(4)));
typedef float v8f __attribute__((ext_vector_type(8)));

#define HDIM 1024
#define LSRC 2048
#define BATCH 16

// fast tanh via hardware exp: tanh(x) = 1 - 2/(e^{2x}+1)
// saturates correctly: x->+inf => 1, x->-inf => -1.  ~5 VALU ops.
__device__ __forceinline__ float fast_tanh(float x) {
    const float e = __expf(2.0f * x);            // v_mul + v_exp_f32
    return 1.0f - __fdividef(2.0f, e + 1.0f);    // v_add + v_rcp + v_mul + v_fma
}

// ---------------------------------------------------------------------------
// out[16 x N] = A[16 x K] * W[N x K]^T + bias    (fp32 WMMA, split-K x8)
// 8 waves per block; wave w accumulates K-range [w*K/8, (w+1)*K/8) with
// V_WMMA_F32_16X16X4_F32, then deterministic LDS reduction of the 8 partial
// 16x16 tiles.  K is a template constant so the inner loop has uniform,
// constant trip count -> compiler unrolls x8 and pipelines the b64 loads
// against the WMMAs (no divergent exec-mask loop).
// Fragment layouts per CDNA5 ISA 7.12.2 (32-bit A 16x4 table):
//   A: lane<16 -> row=lane,    {A[row][k+0], A[row][k+1]}
//      lane>=16-> row=lane-16, {A[row][k+2], A[row][k+3]}
//   B: n = lane&15, same K split by lane half -> contiguous float2 of W[n][kk]
//   C/D: VGPR r -> M = r + 8*(lane>=16), N = lane&15
// ---------------------------------------------------------------------------
template<int K>
__global__ __launch_bounds__(256)
void gemm16_wmma_f32(const float* __restrict__ A, int lda,
                     const float* __restrict__ W, int ldw,
                     const float* __restrict__ bias,
                     float* __restrict__ out, int ldo)
{
    constexpr int KSPAN = K / 8;         // per-wave K range (multiple of 4)
    __shared__ float red[8][16][16];
    const int lane = threadIdx.x;        // 0..31
    const int wv   = threadIdx.y;        // 0..7  (split-K wave id)
    const int n0   = blockIdx.x * 16;
    const int half = lane >> 4;
    const int l16  = lane & 15;

    // fold wave K-offset + lane half-offset into base pointers once
    const float* ap = A + (size_t)l16 * lda        + wv * KSPAN + half * 2;
    const float* wp = W + (size_t)(n0 + l16) * ldw + wv * KSPAN + half * 2;

    v8f acc = {};
#pragma unroll 8
    for (int i = 0; i < KSPAN / 4; ++i) {          // uniform constant bounds
        v2f a = *(const v2f*)(ap + i * 4);
        v2f b = *(const v2f*)(wp + i * 4);
        acc = __builtin_amdgcn_wmma_f32_16x16x4_f32(false, a, false, b,
                                                    (short)0, acc, false, false);
    }
#pragma unroll
    for (int r = 0; r < 8; ++r)
        red[wv][half * 8 + r][l16] = acc[r];
    __syncthreads();

    // 256 threads -> one output element each; fixed-order sum (deterministic)
    const int t   = threadIdx.y * 32 + threadIdx.x;  // 0..255
    const int row = t >> 4;
    const int col = t & 15;
    float s = bias[n0 + col];
#pragma unroll
    for (int w = 0; w < 8; ++w) s += red[w][row][col];
    out[(size_t)row * ldo + n0 + col] = s;
}

// ---------------------------------------------------------------------------
// logits[b][l] = sum_h tanh(qf[b][h] + sf[b][l][h] + cov[b][l]*Wc[h]) * v[h]
// masked rows -> -inf.  One wave32 per (b,l) row; streams states_features
// (134 MB) with non-temporal b128 loads; fast hardware-exp tanh keeps the
// pass memory-bound instead of VALU-bound.
// ---------------------------------------------------------------------------
__global__ __launch_bounds__(256)
void align_logits_kernel(const float* __restrict__ sf,
                         const float* __restrict__ qf,
                         const float* __restrict__ cov,
                         const float* __restrict__ Wc,
                         const float* __restrict__ v,
                         const int*   __restrict__ mask,
                         const int*   __restrict__ is_cov,
                         float* __restrict__ logits)
{
    const int lane = threadIdx.x;                    // 0..31
    const int row  = blockIdx.x * 8 + threadIdx.y;   // (b*LSRC + l)
    if (mask[row] == 0) {
        if (lane == 0) logits[row] = -INFINITY;
        return;
    }
    const int   b    = row >> 11;                    // row / LSRC
    const float cval = (is_cov[0] != 0) ? cov[row] : 0.0f;
    const float* srow = sf + (size_t)row * HDIM;
    const float* qrow = qf + (size_t)b * HDIM;
    float acc = 0.0f;
#pragma unroll
    for (int i = 0; i < 8; ++i) {
        const int h = i * 128 + lane * 4;
        v4f s  = __builtin_nontemporal_load((const v4f*)(srow + h));
        v4f q  = *(const v4f*)(qrow + h);
        v4f wc = *(const v4f*)(Wc + h);
        v4f vv = *(const v4f*)(v + h);
        v4f t  = s + q + cval * wc;
        acc += fast_tanh(t.x) * vv.x + fast_tanh(t.y) * vv.y
             + fast_tanh(t.z) * vv.z + fast_tanh(t.w) * vv.w;
    }
#pragma unroll
    for (int off = 16; off > 0; off >>= 1)
        acc += __shfl_xor(acc, off, 32);
    if (lane == 0) logits[row] = acc;
}

// ---------------------------------------------------------------------------
// Per-batch masked softmax over L; writes align_vectors and new_coverage.
// ---------------------------------------------------------------------------
__global__ __launch_bounds__(256)
void softmax_kernel(const float* __restrict__ logits,
                    const float* __restrict__ cov,
                    float* __restrict__ align_out,
                    float* __restrict__ newcov_out)
{
    __shared__ float red[256];
    const int b = blockIdx.x;
    const int t = threadIdx.x;
    const float* lrow = logits + (size_t)b * LSRC;

    float x[8];
    float mx = -INFINITY;
#pragma unroll
    for (int i = 0; i < 8; ++i) { x[i] = lrow[t + i * 256]; mx = fmaxf(mx, x[i]); }
    red[t] = mx; __syncthreads();
    for (int s = 128; s > 0; s >>= 1) {
        if (t < s) red[t] = fmaxf(red[t], red[t + s]);
        __syncthreads();
    }
    mx = red[0]; __syncthreads();

    float e[8];
    float sum = 0.0f;
#pragma unroll
    for (int i = 0; i < 8; ++i) {
        e[i] = (x[i] == -INFINITY) ? 0.0f : __expf(x[i] - mx);
        sum += e[i];
    }
    red[t] = sum; __syncthreads();
    for (int s = 128; s > 0; s >>= 1) {
        if (t < s) red[t] += red[t + s];
        __syncthreads();
    }
    const float inv = (red[0] > 0.0f) ? (1.0f / red[0]) : 0.0f;
#pragma unroll
    for (int i = 0; i < 8; ++i) {
        const int l = t + i * 256;
        const float a = e[i] * inv;
        align_out[(size_t)b * LSRC + l]  = a;
        newcov_out[(size_t)b * LSRC + l] = cov[(size_t)b * LSRC + l] + a;
    }
}

// ---------------------------------------------------------------------------
// partial[b][chunk][0:H] = sum over 128 l's of align[b][l]*states[b][l][:]
// grid = (16 chunks, B); each thread owns 4 consecutive h -> b128 NT loads
// of the 134 MB states stream.  Fixed reduce order per element.
// ---------------------------------------------------------------------------
__global__ __launch_bounds__(256)
void context_partial_kernel(const float* __restrict__ states,
                            const float* __restrict__ align,
                            float* __restrict__ partial)
{
    const int h4    = threadIdx.x * 4;               // 0..1020
    const int chunk = blockIdx.x;                    // 0..15
    const int b     = blockIdx.y;                    // 0..15
    const float* arow = align + (size_t)b * LSRC + chunk * 128;
    const float* srow = states + ((size_t)b * LSRC + (size_t)chunk * 128) * HDIM + h4;
    v4f acc = {};
#pragma unroll 4
    for (int l = 0; l < 128; ++l) {
        v4f s = __builtin_nontemporal_load((const v4f*)(srow + (size_t)l * HDIM));
        acc += arow[l] * s;
    }
    *(v4f*)(partial + ((size_t)b * 16 + chunk) * HDIM + h4) = acc;
}

// ---------------------------------------------------------------------------
// cat[b][0:H] = sum_chunk partial[b][chunk][:],  cat[b][H:2H] = query[b][:]
// ---------------------------------------------------------------------------
__global__ __launch_bounds__(256)
void build_cat_kernel(const float* __restrict__ partial,
                      const float* __restrict__ query,
                      float* __restrict__ cat)
{
    const int idx = blockIdx.x * 256 + threadIdx.x;  // 0 .. B*HDIM/4
    const int b   = idx >> 8;                        // HDIM/4 = 256 per batch
    const int h4  = (idx & 255) * 4;
    v4f acc = {};
#pragma unroll
    for (int c = 0; c < 16; ++c)
        acc += *(const v4f*)(partial + ((size_t)b * 16 + c) * HDIM + h4);
    *(v4f*)(cat + (size_t)b * (2 * HDIM) + h4)        = acc;
    *(v4f*)(cat + (size_t)b * (2 * HDIM) + HDIM + h4) =
        *(const v4f*)(query + (size_t)b * HDIM + h4);
}

// ---------------------------------------------------------------------------
extern "C" void kernel_launch(void* const* d_in, const int* in_sizes, int n_in,
                              void* d_out, int out_size, void* d_ws, size_t ws_size,
                              hipStream_t stream)
{
    const float* query    = (const float*)d_in[0];   // (16,1,1024)
    const float* states   = (const float*)d_in[1];   // (16,2048,1024)
    const float* sfeat    = (const float*)d_in[2];   // (16,2048,1024)
    const float* coverage = (const float*)d_in[3];   // (16,2048,1)
    const int*   mask     = (const int*)  d_in[4];   // (16,2048)
    const int*   is_cov   = (const int*)  d_in[5];   // scalar
    const float* Wq       = (const float*)d_in[6];   // (1024,1024)
    const float* bq       = (const float*)d_in[7];   // (1024,)
    const float* Wc       = (const float*)d_in[8];   // (1,1024)
    const float* v        = (const float*)d_in[9];   // (1024,1)
    const float* Wout     = (const float*)d_in[10];  // (1024,2048)
    const float* bout     = (const float*)d_in[11];  // (1024,)

    float* out    = (float*)d_out;
    float* attn   = out;                         // 16*1024
    float* newcov = out + BATCH * HDIM;          // 16*2048
    float* align  = newcov + BATCH * LSRC;       // 16*2048

    float* ws      = (float*)d_ws;
    float* qf      = ws;                         // 16*1024
    float* logits  = qf + BATCH * HDIM;          // 16*2048
    float* cat     = logits + BATCH * LSRC;      // 16*2048
    float* partial = cat + BATCH * 2 * HDIM;     // 16*16*1024

    // 1) query_features = query @ Wq.T + bq   (fp32 WMMA, split-K x8)
    gemm16_wmma_f32<HDIM><<<HDIM / 16, dim3(32, 8), 0, stream>>>(
        query, HDIM, Wq, HDIM, bq, qf, HDIM);
    // 2) alignment logits (streams states_features)
    align_logits_kernel<<<(BATCH * LSRC) / 8, dim3(32, 8), 0, stream>>>(
        sfeat, qf, coverage, Wc, v, mask, is_cov, logits);
    // 3) masked softmax -> align_vectors + new_coverage (directly to d_out)
    softmax_kernel<<<BATCH, 256, 0, stream>>>(logits, coverage, align, newcov);
    // 4) context partial sums (streams states)
    context_partial_kernel<<<dim3(16, BATCH), 256, 0, stream>>>(
        states, align, partial);
    // 5) reduce partials + concat with query
    build_cat_kernel<<<(BATCH * HDIM) / (4 * 256), 256, 0, stream>>>(
        partial, query, cat);
    // 6) attn_h = concat(c, query) @ Wout.T + bout  (fp32 WMMA, split-K x8)
    gemm16_wmma_f32<2 * HDIM><<<HDIM / 16, dim3(32, 8), 0, stream>>>(
        cat, 2 * HDIM, Wout, 2 * HDIM, bout, attn, HDIM);
}